// LearnedBFGSSolver_28810640621849
// MI455X (gfx1250) — compile-verified
//
#include <hip/hip_runtime.h>

// Problem constants (match the reference: T=12, B=8, E=32, N=192)
#define T_STEPS 12
#define BE 256                    // B*E independent problems
#define NN 192                    // matrix dimension
#define HPAD 196                  // padded LDS row stride (floats): 16B-aligned rows,
                                  // 8*HPAD % 64 == 32 -> conflict-free C-fragment loads
#define NT 12                     // NN/16 tiles per dimension
#define NTILES (NT*NT)            // 144 tiles
#define NTHREADS 256
#define NWAVES 8
#define TILES_PER_WAVE (NTILES/NWAVES)  // 18

typedef __attribute__((ext_vector_type(2))) float v2f;
typedef __attribute__((ext_vector_type(4))) float v4f;
typedef __attribute__((ext_vector_type(8))) float v8f;

__global__ __launch_bounds__(NTHREADS)
void bfgs_rank2_wmma_kernel(const float* __restrict__ H0,      // [NN,NN] shared inverse-Hessian init
                            const float* __restrict__ steps,   // [T,BE,NN]
                            const float* __restrict__ dgrads,  // [T,BE,NN]
                            float* __restrict__ out)           // [BE,NN,NN]
{
    extern __shared__ float lds[];
    float* H   = lds;                  // NN * HPAD  (H resident for all 12 steps)
    float* sv  = H   + NN * HPAD;      // s vector
    float* yv  = sv  + NN;             // y vector
    float* Hy  = yv  + NN;             // H @ y
    float* yH  = Hy  + NN;             // y^T @ H
    float* u1  = yH  + NN;             // first rank-1 left factor
    float* v2  = u1  + NN;             // second rank-1 right factor
    float* red = v2  + NN;             // [0]=sdot, [1]=yHy

    const int tid  = threadIdx.x;
    const int p    = blockIdx.x;       // flat (b*E + e) problem index
    const int lane = tid & 31;
    const int wave = tid >> 5;
    const int li   = lane & 15;
    const float hmask = (lane < 16) ? 1.0f : 0.0f;   // zero K=2,3 lanes of A/B frags

    // ---- load shared H0 into LDS (padded rows), vectorized 16B ----
    for (int q = tid; q < (NN * NN) / 4; q += NTHREADS) {
        int r = q / (NN / 4);
        int c = (q % (NN / 4)) * 4;
        *(v4f*)(H + r * HPAD + c) = ((const v4f*)H0)[q];
    }
    __syncthreads();

    for (int t = 0; t < T_STEPS; ++t) {
        const float* s_t = steps  + (size_t)(t * BE + p) * NN;
        const float* y_t = dgrads + (size_t)(t * BE + p) * NN;

        if (tid < NN) {
            sv[tid] = s_t[tid];
            yv[tid] = y_t[tid];
            if (t + 1 < T_STEPS) {   // global_prefetch_b8 of next step's vectors
                __builtin_prefetch(s_t + (size_t)BE * NN + tid, 0, 3);
                __builtin_prefetch(y_t + (size_t)BE * NN + tid, 0, 3);
            }
        }
        __syncthreads();

        // ---- matvecs: Hy[n] = H[n,:].y  (row dot), yH[m] = y.H[:,m] (col dot) ----
        if (tid < NN) {
            const float* row = H + tid * HPAD;
            float acc = 0.0f;
            #pragma unroll 8
            for (int m = 0; m < NN; ++m) acc += row[m] * yv[m];
            Hy[tid] = acc;
            float acc2 = 0.0f;
            #pragma unroll 8
            for (int n = 0; n < NN; ++n) acc2 += yv[n] * H[n * HPAD + tid];
            yH[tid] = acc2;
        }
        __syncthreads();

        // ---- deterministic scalar reductions ----
        if (tid == 0) {
            float sdot = 0.0f, yhy = 0.0f;
            for (int n = 0; n < NN; ++n) { sdot += sv[n] * yv[n]; yhy += yv[n] * Hy[n]; }
            red[0] = sdot; red[1] = yhy;
        }
        __syncthreads();

        const float sdot = red[0];
        const float yhy  = red[1];
        const float inv  = (sdot != 0.0f) ? 1.0f / sdot : 0.0f;              // mask: sdot==0 -> delta==0
        const float c1   = (sdot != 0.0f) ? (sdot + yhy) / (sdot * sdot) : 0.0f;
        if (tid < NN) {
            u1[tid] = c1 * sv[tid] - Hy[tid] * inv;
            v2[tid] = -yH[tid] * inv;
        }
        __syncthreads();

        // ---- rank-2 update as K=2 (zero-padded to K=4) fp32 WMMA per 16x16 tile ----
        // H_tile += [u1 | s | 0 | 0](16x4) x [s ; v2 ; 0 ; 0](4x16), C = H_tile
        for (int tt = 0; tt < TILES_PER_WAVE; ++tt) {
            const int tile = wave * TILES_PER_WAVE + tt;    // wave-uniform -> EXEC stays all-1s
            const int R = (tile / NT) * 16;
            const int C = (tile % NT) * 16;

            // A frag (16x4 f32, 2 VGPRs): lanes 0-15 hold K=0,1 ; lanes 16-31 hold K=2,3 (zero)
            v2f a; a.x = u1[R + li] * hmask; a.y = sv[R + li] * hmask;
            // B frag (4x16 f32, 2 VGPRs): mirrored K split across lane halves
            v2f b; b.x = sv[C + li] * hmask; b.y = v2[C + li] * hmask;

            // C/D frag (16x16 f32, 8 VGPRs): lanes 0-15 rows R..R+7, lanes 16-31 rows R+8..R+15
            const int col = C + li;
            const int rb  = R + ((lane >> 4) << 3);
            v8f cf;
            #pragma unroll
            for (int v = 0; v < 8; ++v) cf[v] = H[(rb + v) * HPAD + col];

            v8f d = __builtin_amdgcn_wmma_f32_16x16x4_f32(
                false, a, false, b, (short)0, cf, false, false);

            #pragma unroll
            for (int v = 0; v < 8; ++v) H[(rb + v) * HPAD + col] = d[v];
        }
        __syncthreads();
    }

    // ---- write H_final for this problem, vectorized 16B ----
    float* outp = out + (size_t)p * NN * NN;
    for (int q = tid; q < (NN * NN) / 4; q += NTHREADS) {
        int r = q / (NN / 4);
        int c = (q % (NN / 4)) * 4;
        *(v4f*)(outp + r * NN + c) = *(const v4f*)(H + r * HPAD + c);
    }
}

extern "C" void kernel_launch(void* const* d_in, const int* in_sizes, int n_in,
                              void* d_out, int out_size, void* d_ws, size_t ws_size,
                              hipStream_t stream) {
    (void)in_sizes; (void)n_in; (void)out_size; (void)d_ws; (void)ws_size;
    const float* H0     = (const float*)d_in[0];   // inv_hessian [192,192] f32
    const float* steps  = (const float*)d_in[1];   // steps       [12,8,32,192] f32
    const float* dgrads = (const float*)d_in[2];   // delta_grads [12,8,32,192] f32
    float* out          = (float*)d_out;           // [8,32,192,192] f32

    const size_t shmem = (size_t)(NN * HPAD + 6 * NN + 8) * sizeof(float);  // ~151.5 KB < 320 KB
    hipFuncSetAttribute(reinterpret_cast<const void*>(bfgs_rank2_wmma_kernel),
                        hipFuncAttributeMaxDynamicSharedMemorySize, (int)shmem);
    bfgs_rank2_wmma_kernel<<<BE, NTHREADS, shmem, stream>>>(H0, steps, dgrads, out);
}